// Forward_63539746177538
// MI455X (gfx1250) — compile-verified
//
#include <hip/hip_runtime.h>
#include <math.h>

typedef __attribute__((ext_vector_type(2))) float v2f;
typedef __attribute__((ext_vector_type(8))) float v8f;

#define BM 128          // block tile rows
#define BN 64           // block tile cols
#define BK 32           // K slice
#define STRIDE_A 290    // floats per LDS pair-row of A (128*2 + 34), stride%64==34
#define STRIDE_B 162    // floats per LDS pair-row of B ( 64*2 + 34), stride%64==34

// ---------------------------------------------------------------------------
// GEMM1: Y1[m][n] = sum_k G[m][k] * w1[n][k] + b1[n]   (M=2048, N=1024, K=8)
// ---------------------------------------------------------------------------
__global__ __launch_bounds__(256) void gemm1_kernel(
    const float* __restrict__ G, const float* __restrict__ w1,
    const float* __restrict__ b1, float* __restrict__ Y1)
{
    int n = blockIdx.x * 256 + threadIdx.x;   // 0..1023
    int m = blockIdx.y;                       // 0..2047
    float g[8];
#pragma unroll
    for (int i = 0; i < 8; ++i) g[i] = G[m * 8 + i];
    float acc = b1[n];
#pragma unroll
    for (int i = 0; i < 8; ++i) acc += g[i] * w1[n * 8 + i];
    Y1[(size_t)m * 1024 + n] = acc;
}

// ---------------------------------------------------------------------------
// Per-feature batch-norm stats over M rows:
//   scale[c] = gamma[c]/sqrt(var+eps),  shift[c] = beta[c] - scale[c]*mean
// ---------------------------------------------------------------------------
__global__ __launch_bounds__(256) void bn_stats_kernel(
    const float* __restrict__ Y, int M, int N,
    const float* __restrict__ gamma, const float* __restrict__ beta,
    float* __restrict__ scale, float* __restrict__ shift)
{
    __shared__ float ssum[256];
    __shared__ float ssq[256];
    int c = blockIdx.x;
    float s = 0.f, q = 0.f;
    for (int m = threadIdx.x; m < M; m += 256) {
        float v = Y[(size_t)m * N + c];
        s += v;
        q += v * v;
    }
    ssum[threadIdx.x] = s;
    ssq[threadIdx.x]  = q;
    __syncthreads();
    for (int off = 128; off > 0; off >>= 1) {
        if (threadIdx.x < off) {
            ssum[threadIdx.x] += ssum[threadIdx.x + off];
            ssq[threadIdx.x]  += ssq[threadIdx.x + off];
        }
        __syncthreads();
    }
    if (threadIdx.x == 0) {
        float invM  = 1.0f / (float)M;
        float mean  = ssum[0] * invM;
        float var   = ssq[0] * invM - mean * mean;
        float a     = gamma[c] * rsqrtf(var + 1e-5f);
        scale[c] = a;
        shift[c] = beta[c] - a * mean;
    }
}

// ---------------------------------------------------------------------------
// WMMA GEMM with fused BN+ReLU on A:
//   Y[m][n] = sum_k relu(scaleA[k]*A[m][k]+shiftA[k]) * W[n][k] + bias[n]
// fp32 V_WMMA_F32_16X16X4_F32. 256 threads = 8 waves; block tile 128x64;
// each wave owns a 32x32 tile (4 accumulators) -> 1 ds_load_b64 per WMMA.
// LDS stores K pair-interleaved: element (k,m) at [k>>1][2m + (k&1)], so a
// lane's (K, K+1) fragment pair is one aligned b64 load (no repack movs).
// Row strides are 34 mod 64 -> conflict-free writes, ~conflict-free reads.
// M must be a multiple of 128 (M=2048). N guarded (N=150 case).
// ---------------------------------------------------------------------------
__global__ __launch_bounds__(256) void gemm_bn_wmma_kernel(
    const float* __restrict__ A,      // M x K (pre-BN)
    const float* __restrict__ scaleA, // K
    const float* __restrict__ shiftA, // K
    const float* __restrict__ W,      // N x K (row-major, w[n][k])
    const float* __restrict__ bias,   // N
    float* __restrict__ Y,            // M x N
    int M, int N, int K)
{
    __shared__ float As[(BK / 2) * STRIDE_A];   // 18560 B
    __shared__ float Bs[(BK / 2) * STRIDE_B];   // 10368 B

    const int tid  = threadIdx.x;
    const int lane = tid & 31;
    const int wave = tid >> 5;
    const int wm   = wave >> 1;  // 0..3 : 32-row slab
    const int wn   = wave & 1;   // 0..1 : 32-col slab
    const int mBase = blockIdx.y * BM;
    const int nBase = blockIdx.x * BN;

    const int kcol = tid & 31;   // k within K-slice (cooperative loads)
    const int rrow = tid >> 5;   // 0..7
    // interleaved write offset pieces for the loader
    const int wpairA = (kcol >> 1) * STRIDE_A + (kcol & 1);
    const int wpairB = (kcol >> 1) * STRIDE_B + (kcol & 1);

    v8f acc00 = {}, acc01 = {}, acc10 = {}, acc11 = {};

    // per-lane fragment base offsets (floats)
    const int half = lane >> 4;          // 0 or 1 -> owns K pair p, p+1
    const int lm   = lane & 15;
    const int aOff = half * STRIDE_A + (wm * 32 + lm) * 2;
    const int bOff = half * STRIDE_B + (wn * 32 + lm) * 2;

    for (int kb = 0; kb < K; kb += BK) {
        // A panel (128 x 32), bn_relu fused, coalesced along k
        const float sc = scaleA[kb + kcol];
        const float sh = shiftA[kb + kcol];
#pragma unroll
        for (int i = 0; i < 16; ++i) {
            int m = rrow + i * 8;
            float v = A[(size_t)(mBase + m) * K + kb + kcol];
            v = sc * v + sh;
            As[wpairA + m * 2] = v > 0.f ? v : 0.f;
        }
        // B panel (32 x 64): Bs(k,n) = W[n][k], zero-pad n >= N
#pragma unroll
        for (int i = 0; i < 8; ++i) {
            int n  = rrow + i * 8;
            int ng = nBase + n;
            Bs[wpairB + n * 2] = (ng < N) ? W[(size_t)ng * K + kb + kcol] : 0.f;
        }
        __syncthreads();

#pragma unroll
        for (int s = 0; s < 8; ++s) {    // dk = 4*s
            v2f a0 = *(const v2f*)&As[(2 * s) * STRIDE_A + aOff];
            v2f a1 = *(const v2f*)&As[(2 * s) * STRIDE_A + aOff + 32];
            v2f b0 = *(const v2f*)&Bs[(2 * s) * STRIDE_B + bOff];
            v2f b1 = *(const v2f*)&Bs[(2 * s) * STRIDE_B + bOff + 32];
            acc00 = __builtin_amdgcn_wmma_f32_16x16x4_f32(
                        false, a0, false, b0, (short)0, acc00, false, false);
            acc01 = __builtin_amdgcn_wmma_f32_16x16x4_f32(
                        false, a0, false, b1, (short)0, acc01, false, false);
            acc10 = __builtin_amdgcn_wmma_f32_16x16x4_f32(
                        false, a1, false, b0, (short)0, acc10, false, false);
            acc11 = __builtin_amdgcn_wmma_f32_16x16x4_f32(
                        false, a1, false, b1, (short)0, acc11, false, false);
        }
        __syncthreads();
    }

    // Epilogue. C/D layout: VGPR r -> (M = r + 8*(lane>>4), N = lane&15)
    const int mh = half * 8;
    const int n0 = nBase + wn * 32 + lm;
    const int n1 = n0 + 16;
    const float bias0 = (n0 < N) ? bias[n0] : 0.f;
    const float bias1 = (n1 < N) ? bias[n1] : 0.f;
#pragma unroll
    for (int r = 0; r < 8; ++r) {
        int mg0 = mBase + wm * 32 + mh + r;       // rows of acc*0 / acc*1 slab 0
        int mg1 = mg0 + 16;                        // slab 1
        if (n0 < N) {
            Y[(size_t)mg0 * N + n0] = acc00[r] + bias0;
            Y[(size_t)mg1 * N + n0] = acc10[r] + bias0;
        }
        if (n1 < N) {
            Y[(size_t)mg0 * N + n1] = acc01[r] + bias1;
            Y[(size_t)mg1 * N + n1] = acc11[r] + bias1;
        }
    }
}

// ---------------------------------------------------------------------------
// Fused per-row tail: bn_relu(layer3) -> ct1(k8,s2,p3) -> ct2(k5,s1,p2)
// -> ct3(k5,s1,p2) -> 1x1 conv -> Lorentz sum. One block per batch row.
// ---------------------------------------------------------------------------
__global__ __launch_bounds__(256) void conv_lorentz_kernel(
    const float* __restrict__ Y3,
    const float* __restrict__ scale3, const float* __restrict__ shift3,
    const float* __restrict__ ct1_w, const float* __restrict__ ct1_b,
    const float* __restrict__ ct2_w, const float* __restrict__ ct2_b,
    const float* __restrict__ ct3_w, const float* __restrict__ ct3_b,
    const float* __restrict__ cf_w,  const float* __restrict__ cf_b,
    float* __restrict__ out)
{
    __shared__ float s0[150];
    __shared__ float xa[4][300];
    __shared__ float xb[4][300];
    __shared__ float S[300];

    const int row = blockIdx.x;
    const int tid = threadIdx.x;

    for (int j = tid; j < 150; j += 256) {
        float v = Y3[(size_t)row * 150 + j];
        v = scale3[j] * v + shift3[j];
        s0[j] = v > 0.f ? v : 0.f;
    }
    __syncthreads();

    // ct1: 1->4 ch, k=8, stride=2, pad=3 ; y[co][j]=b+sum_t s0[t]*w[0][co][j+3-2t]
    for (int idx = tid; idx < 1200; idx += 256) {
        int co = idx / 300, j = idx - co * 300;
        float acc = ct1_b[co];
        int tmin = (j >= 4) ? ((j - 3) >> 1) : 0;
        int tmax = (j + 3) >> 1; if (tmax > 149) tmax = 149;
        for (int t = tmin; t <= tmax; ++t) {
            int q = j + 3 - 2 * t;
            acc += s0[t] * ct1_w[co * 8 + q];
        }
        xa[co][j] = acc;
    }
    __syncthreads();

    // ct2: 4->4, k=5, stride=1, pad=2
    for (int idx = tid; idx < 1200; idx += 256) {
        int co = idx / 300, j = idx - co * 300;
        float acc = ct2_b[co];
#pragma unroll
        for (int ci = 0; ci < 4; ++ci) {
#pragma unroll
            for (int q = 0; q < 5; ++q) {
                int t = j + 2 - q;
                if (t >= 0 && t < 300)
                    acc += xa[ci][t] * ct2_w[(ci * 4 + co) * 5 + q];
            }
        }
        xb[co][j] = acc;
    }
    __syncthreads();

    // ct3: 4->4, k=5, stride=1, pad=2
    for (int idx = tid; idx < 1200; idx += 256) {
        int co = idx / 300, j = idx - co * 300;
        float acc = ct3_b[co];
#pragma unroll
        for (int ci = 0; ci < 4; ++ci) {
#pragma unroll
            for (int q = 0; q < 5; ++q) {
                int t = j + 2 - q;
                if (t >= 0 && t < 300)
                    acc += xb[ci][t] * ct3_w[(ci * 4 + co) * 5 + q];
            }
        }
        xa[co][j] = acc;
    }
    __syncthreads();

    // 1x1 conv: 4 -> 1
    for (int j = tid; j < 300; j += 256) {
        float acc = cf_b[0];
#pragma unroll
        for (int ci = 0; ci < 4; ++ci) acc += xa[ci][j] * cf_w[ci];
        S[j] = acc;
    }
    __syncthreads();

    // Lorentz: 300 frequencies x 100 oscillators (LDS broadcast reads)
    for (int j = tid; j < 300; j += 256) {
        float w = 0.8f + (float)j / 300.0f;
        float accT = 0.f;
        for (int i = 0; i < 100; ++i) {
            float wp   = S[3 * i];
            float wp2  = wp * wp;
            float w0   = S[3 * i + 1];
            float g    = S[3 * i + 2];
            float sub1 = w0 * w0 - w * w;
            float wg   = w * g;
            float den  = sub1 * sub1 + wg * wg;
            float inv  = 1.0f / den;
            float e1   = wp2 * sub1 * inv;
            float e2   = wp2 * wg * inv;
            float m    = 0.5f * (e1 * e1 + e2 * e2);
            float hn   = m + 0.5f * e1; hn   = hn   > 0.f ? hn   : 0.f;
            float kap2 = m - 0.5f * e1; kap2 = kap2 > 0.f ? kap2 : 0.f;
            float nn   = sqrtf(hn);
            float np1  = nn + 1.0f;
            accT += 4.0f * nn / (np1 * np1 + kap2);
        }
        out[(size_t)row * 300 + j] = accT;
    }
}

// ---------------------------------------------------------------------------
extern "C" void kernel_launch(void* const* d_in, const int* in_sizes, int n_in,
                              void* d_out, int out_size, void* d_ws, size_t ws_size,
                              hipStream_t stream)
{
    const float* G    = (const float*)d_in[0];
    const float* w1   = (const float*)d_in[1];
    const float* b1   = (const float*)d_in[2];
    const float* g1   = (const float*)d_in[3];
    const float* be1  = (const float*)d_in[4];
    const float* w2   = (const float*)d_in[5];
    const float* b2   = (const float*)d_in[6];
    const float* g2   = (const float*)d_in[7];
    const float* be2  = (const float*)d_in[8];
    const float* w3   = (const float*)d_in[9];
    const float* b3   = (const float*)d_in[10];
    const float* g3   = (const float*)d_in[11];
    const float* be3  = (const float*)d_in[12];
    const float* ct1w = (const float*)d_in[13];
    const float* ct1b = (const float*)d_in[14];
    const float* ct2w = (const float*)d_in[15];
    const float* ct2b = (const float*)d_in[16];
    const float* ct3w = (const float*)d_in[17];
    const float* ct3b = (const float*)d_in[18];
    const float* cfw  = (const float*)d_in[19];
    const float* cfb  = (const float*)d_in[20];
    float* out = (float*)d_out;

    const int B = 2048, H = 1024, N3 = 150;

    // Workspace carve-up (floats): ~18 MB total
    float* Y1 = (float*)d_ws;                 // B*H
    float* Y2 = Y1 + (size_t)B * H;           // B*H
    float* Y3 = Y2 + (size_t)B * H;           // B*N3
    float* s1 = Y3 + (size_t)B * N3;          // H
    float* h1 = s1 + H;                       // H
    float* s2 = h1 + H;                       // H
    float* h2 = s2 + H;                       // H
    float* s3 = h2 + H;                       // 256 (150 used)
    float* h3 = s3 + 256;                     // 256 (150 used)

    // Layer 1 (K=8): plain kernel + stats
    gemm1_kernel<<<dim3(H / 256, B), 256, 0, stream>>>(G, w1, b1, Y1);
    bn_stats_kernel<<<H, 256, 0, stream>>>(Y1, B, H, g1, be1, s1, h1);

    // Layer 2: WMMA GEMM with fused bn_relu(Y1) input
    gemm_bn_wmma_kernel<<<dim3(H / BN, B / BM), 256, 0, stream>>>(
        Y1, s1, h1, w2, b2, Y2, B, H, H);
    bn_stats_kernel<<<H, 256, 0, stream>>>(Y2, B, H, g2, be2, s2, h2);

    // Layer 3: WMMA GEMM, N=150 (3 column tiles, guarded)
    gemm_bn_wmma_kernel<<<dim3((N3 + BN - 1) / BN, B / BM), 256, 0, stream>>>(
        Y2, s2, h2, w3, b3, Y3, B, N3, H);
    bn_stats_kernel<<<N3, 256, 0, stream>>>(Y3, B, N3, g3, be3, s3, h3);

    // Fused conv-transpose chain + Lorentz, one block per batch row
    conv_lorentz_kernel<<<B, 256, 0, stream>>>(
        Y3, s3, h3, ct1w, ct1b, ct2w, ct2b, ct3w, ct3b, cfw, cfb, out);
}